// SparsemaxBisectLoss_48180943126720
// MI455X (gfx1250) — compile-verified
//
#include <hip/hip_runtime.h>

// ---------------------------------------------------------------------------
// SparsemaxBisectLoss for MI455X (gfx1250, wave32)
//   X, target : (4096, 32000) f32     out : scalar f32
//
// Strategy: register-resident row (100 f32/thread @ 320 thr), 50 bisection
// passes entirely from VGPRs, target row streamed into LDS via the CDNA5
// async global->LDS engine (ASYNCcnt) overlapped with the bisection,
// deterministic two-kernel reduction (no float atomics).
// ---------------------------------------------------------------------------

#define C_DIM   32000
#define N_ROWS  4096
#define TPB     320          // 10 wave32 waves; 320 * 25 float4 = 32000 floats
#define WPB     (TPB / 32)
#define VPT     25           // float4 per thread
#define N_ITER  50
#define RED_PAD 16           // LDS floats reserved for reductions (16B-aligns tgt)

#ifndef __has_builtin
#define __has_builtin(x) 0
#endif

#if __has_builtin(__builtin_amdgcn_global_load_async_to_lds_b128)
#define HAVE_ASYNC 1
#else
#define HAVE_ASYNC 0
#endif

// b128 async copy operand types: v4i pointers in AS1 (global) / AS3 (LDS).
typedef int v4i __attribute__((ext_vector_type(4)));
typedef __attribute__((address_space(1))) v4i* gptr_b128;
typedef __attribute__((address_space(3))) v4i* lptr_b128;

// Address-space casts via integer round-trip (always type-checks in HIP).
#define TO_GLOBAL_B128(p) ((gptr_b128)(unsigned long long)(const void*)(p))
#define TO_LDS_B128(p)    ((lptr_b128)(unsigned int)(unsigned long long)(const void*)(p))

__device__ __forceinline__ float wave_sum(float v) {
#pragma unroll
  for (int m = 16; m > 0; m >>= 1) v += __shfl_xor(v, m, 32);
  return v;
}

__device__ __forceinline__ float wave_max(float v) {
#pragma unroll
  for (int m = 16; m > 0; m >>= 1) v = fmaxf(v, __shfl_xor(v, m, 32));
  return v;
}

// Block reduction across 10 waves; every thread gets the result.
__device__ __forceinline__ float block_sum(float v, float* red) {
  v = wave_sum(v);
  const int wid = threadIdx.x >> 5;
  if ((threadIdx.x & 31) == 0) red[wid] = v;
  __syncthreads();
  float r = red[0];
#pragma unroll
  for (int i = 1; i < WPB; ++i) r += red[i];
  __syncthreads();
  return r;
}

__device__ __forceinline__ float block_max(float v, float* red) {
  v = wave_max(v);
  const int wid = threadIdx.x >> 5;
  if ((threadIdx.x & 31) == 0) red[wid] = v;
  __syncthreads();
  float r = red[0];
#pragma unroll
  for (int i = 1; i < WPB; ++i) r = fmaxf(r, red[i]);
  __syncthreads();
  return r;
}

__global__ void __launch_bounds__(TPB)
sparsemax_row_loss_kernel(const float* __restrict__ X,
                          const float* __restrict__ T,
                          float* __restrict__ row_loss) {
  extern __shared__ float smem[];
  float* red = smem;            // RED_PAD floats for reductions
  float* tgt = smem + RED_PAD;  // C_DIM floats: async-staged target row

  const int tid = threadIdx.x;
  const int row = blockIdx.x;

  const float4* Xr = (const float4*)(X + (size_t)row * C_DIM);
  const float4* Tr = (const float4*)(T + (size_t)row * C_DIM);

  // ---- load this row of X into registers (coalesced float4) ----
  float4 xv[VPT];
#pragma unroll
  for (int i = 0; i < VPT; ++i) xv[i] = Xr[i * TPB + tid];

  // ---- kick off target row -> LDS, overlapped with the whole bisection ----
#if HAVE_ASYNC
#pragma unroll
  for (int i = 0; i < VPT; ++i) {
    const int idx = i * TPB + tid;
    __builtin_amdgcn_global_load_async_to_lds_b128(
        TO_GLOBAL_B128(Tr + idx), TO_LDS_B128(tgt + 4 * idx), 0, 0);
  }
#else
  // Fallback: pull the target row toward L2 (global_prefetch_b8) so the tail
  // pass hits cache after ~50 bisection passes of latency to hide it behind.
  for (int i = tid; i < (C_DIM * 4) / 128; i += TPB)
    __builtin_prefetch(((const char*)Tr) + (size_t)i * 128, 0, 1);
#endif

  // ---- row max ----
  float m = -INFINITY;
#pragma unroll
  for (int i = 0; i < VPT; ++i) {
    m = fmaxf(m, fmaxf(fmaxf(xv[i].x, xv[i].y), fmaxf(xv[i].z, xv[i].w)));
  }
  m = block_max(m, red);

  // ---- bisection setup (mirrors the reference exactly) ----
  float tau_lo = m - 1.0f;
  float dm     = 1.0f - 1.0f / (float)C_DIM;   // tau_hi - tau_lo

  float part = 0.0f;
#pragma unroll
  for (int i = 0; i < VPT; ++i) {
    part += fmaxf(xv[i].x - tau_lo, 0.0f) + fmaxf(xv[i].y - tau_lo, 0.0f) +
            fmaxf(xv[i].z - tau_lo, 0.0f) + fmaxf(xv[i].w - tau_lo, 0.0f);
  }
  const float f_lo = block_sum(part, red) - 1.0f;

  // ---- 50 register-resident relu-sum passes ----
  float tau_m = tau_lo;
#pragma unroll 1
  for (int it = 0; it < N_ITER; ++it) {
    dm *= 0.5f;
    tau_m = tau_lo + dm;
    float s = 0.0f;
#pragma unroll
    for (int i = 0; i < VPT; ++i) {
      s += fmaxf(xv[i].x - tau_m, 0.0f) + fmaxf(xv[i].y - tau_m, 0.0f) +
           fmaxf(xv[i].z - tau_m, 0.0f) + fmaxf(xv[i].w - tau_m, 0.0f);
    }
    const float f_m = block_sum(s, red) - 1.0f;   // uniform across block
    if (f_m * f_lo >= 0.0f) tau_lo = tau_m;       // uniform branchless select
  }

  // ---- wait for the async-staged target row ----
#if HAVE_ASYNC
#if __has_builtin(__builtin_amdgcn_s_wait_asynccnt)
  __builtin_amdgcn_s_wait_asynccnt(0);
#else
  asm volatile("s_wait_asynccnt 0" ::: "memory");
#endif
#endif
  __syncthreads();

  // ---- fused tail: S = sum q, Q2 = sum q^2, QX = sum q*x, TX = sum t*x ----
  float S = 0.0f, Q2 = 0.0f, QX = 0.0f, TX = 0.0f;
#pragma unroll
  for (int i = 0; i < VPT; ++i) {
    const int idx = i * TPB + tid;
    const float4 x4 = xv[i];
#if HAVE_ASYNC
    const float4 t4 = *(const float4*)(tgt + 4 * idx);   // ds_load_b128
#else
    const float4 t4 = Tr[idx];
#endif
    { float q = fmaxf(x4.x - tau_m, 0.0f); S += q; Q2 += q * q; QX += q * x4.x; TX += t4.x * x4.x; }
    { float q = fmaxf(x4.y - tau_m, 0.0f); S += q; Q2 += q * q; QX += q * x4.y; TX += t4.y * x4.y; }
    { float q = fmaxf(x4.z - tau_m, 0.0f); S += q; Q2 += q * q; QX += q * x4.z; TX += t4.z * x4.z; }
    { float q = fmaxf(x4.w - tau_m, 0.0f); S += q; Q2 += q * q; QX += q * x4.w; TX += t4.w * x4.w; }
  }
  S  = block_sum(S,  red);
  Q2 = block_sum(Q2, red);
  QX = block_sum(QX, red);
  TX = block_sum(TX, red);

  if (tid == 0) {
    // p = q/S ; loss = 0.5*(1 - Q2/S^2) + QX/S - TX
    const float inv = 1.0f / S;
    row_loss[row] = 0.5f * (1.0f - Q2 * inv * inv) + QX * inv - TX;
  }
}

// Deterministic final reduction: mean over 4096 row losses.
__global__ void __launch_bounds__(256)
sparsemax_finalize_kernel(const float* __restrict__ row_loss, float* __restrict__ out) {
  __shared__ float s[8];
  float v = 0.0f;
  for (int i = threadIdx.x; i < N_ROWS; i += 256) v += row_loss[i];
  v = wave_sum(v);
  if ((threadIdx.x & 31) == 0) s[threadIdx.x >> 5] = v;
  __syncthreads();
  if (threadIdx.x == 0) {
    float r = 0.0f;
#pragma unroll
    for (int i = 0; i < 8; ++i) r += s[i];
    out[0] = r * (1.0f / (float)N_ROWS);
  }
}

extern "C" void kernel_launch(void* const* d_in, const int* in_sizes, int n_in,
                              void* d_out, int out_size, void* d_ws, size_t ws_size,
                              hipStream_t stream) {
  (void)in_sizes; (void)n_in; (void)out_size; (void)ws_size;
  const float* X      = (const float*)d_in[0];
  const float* target = (const float*)d_in[1];
  float* out = (float*)d_out;
  float* ws  = (float*)d_ws;          // 4096 per-row losses

  const size_t shmem = (size_t)(RED_PAD + C_DIM) * sizeof(float);  // ~128 KB; WGP has 320 KB
  sparsemax_row_loss_kernel<<<N_ROWS, TPB, shmem, stream>>>(X, target, ws);
  sparsemax_finalize_kernel<<<1, 256, 0, stream>>>(ws, out);
}